// MultiModalEncoder_14800457302120
// MI455X (gfx1250) — compile-verified
//
#include <hip/hip_runtime.h>

typedef __attribute__((ext_vector_type(2))) float v2f;
typedef __attribute__((ext_vector_type(8))) float v8f;

#define BATCH   8
#define T_LEN   262144
#define EMBED   128
#define PATCH   64
#define KW      5
#define PADL    2
#define SEQ     (T_LEN / 2)          // 131072 output positions per batch
#define KSTEPS  (EMBED / 4)          // 32 wmma K-steps
#define NTILES  (PATCH / 16)         // 4 N-tiles of 16

// Fused conv1d(1->128,k5,s2,p2) + bias + relu + linear(128->64) + bias.
// One wave computes a 16-position x 64-output tile via v_wmma_f32_16x16x4_f32.
__global__ __launch_bounds__(256) void fused_conv_gemm(
    const float* __restrict__ x,       // [B, T]
    const float* __restrict__ conv_w,  // [128, 1, 5]
    const float* __restrict__ conv_b,  // [128]
    const float* __restrict__ lin_w,   // [64, 128]
    const float* __restrict__ lin_b,   // [64]
    float* __restrict__ out)           // [B, SEQ, 64]
{
    // B matrix (lin_w^T) staged in LDS in exact WMMA fragment layout:
    // sB[((j*4 + t)*32 + lane)*2 + p] = B[4j + (lane<16?0:2) + p][16t + (lane&15)]
    __shared__ __align__(16) float sB[KSTEPS * NTILES * 32 * 2];  // 8192 floats, 32 KB
    __shared__ __align__(16) float sW[EMBED * 8];                 // [ch][w0..w4, bias, 0, 0]
    __shared__ float sLB[PATCH];

    const int tid = threadIdx.x;

    // ---- stage conv weights + bias (padded to 8 floats/channel) ----
    for (int i = tid; i < EMBED * 8; i += 256) {
        int ch = i >> 3, t = i & 7;
        float v = 0.0f;
        if (t < KW)       v = conv_w[ch * KW + t];
        else if (t == KW) v = conv_b[ch];
        sW[i] = v;
    }
    // ---- stage B fragments (transpose of lin_w into fragment order) ----
    for (int i = tid; i < KSTEPS * NTILES * 32 * 2; i += 256) {
        int j  = i >> 8;        // k-step       (256 floats per k-step)
        int r  = i & 255;
        int t  = r >> 6;        // n-tile       (64 floats per tile)
        int r2 = r & 63;
        int l  = r2 >> 1;       // lane
        int p  = r2 & 1;        // element within v2f
        int k  = 4 * j + ((l & 16) >> 3) + p;   // lanes>=16 hold K rows +2
        int n  = 16 * t + (l & 15);
        sB[i] = lin_w[n * EMBED + k];
    }
    if (tid < PATCH) sLB[tid] = lin_b[tid];
    __syncthreads();

    const int lane = tid & 31;
    const int wave = tid >> 5;
    const int tile = blockIdx.x * 8 + wave;     // 65536 tiles of 16 rows
    const int m0   = tile * 16;                 // flattened (b*SEQ + s) base
    const int b    = m0 / SEQ;                  // SEQ is 16-aligned: no tile crosses batches
    const int s0   = m0 - b * SEQ;
    const int s    = s0 + (lane & 15);          // this lane's sequence position

    // ---- per-lane input window (stride 2, left pad 2), boundary-predicated ----
    const float* xb = x + (long)b * T_LEN;
    float xs[KW];
#pragma unroll
    for (int t = 0; t < KW; ++t) {
        int j = 2 * s + t - PADL;
        xs[t] = (j >= 0 && j < T_LEN) ? xb[j] : 0.0f;
    }

    const int half = (lane & 16) >> 3;          // lanes>=16 compute channels +2
    v8f acc0 = {}, acc1 = {}, acc2 = {}, acc3 = {};

#pragma unroll 4
    for (int j = 0; j < KSTEPS; ++j) {
        // --- conv + bias + relu for this lane's two channels -> A fragment ---
        const float* w = &sW[(4 * j + half) * 8];
        float a0 = w[5];       // bias ch0
        float a1 = w[13];      // bias ch1
#pragma unroll
        for (int t = 0; t < KW; ++t) {
            a0 = fmaf(w[t],     xs[t], a0);
            a1 = fmaf(w[8 + t], xs[t], a1);
        }
        a0 = fmaxf(a0, 0.0f);
        a1 = fmaxf(a1, 0.0f);
        v2f A = {a0, a1};

        // --- B fragments: conflict-free 8B LDS loads in fragment layout ---
        const v2f* bf = ((const v2f*)sB) + j * (NTILES * 32) + lane;
        v2f B0 = bf[0];
        v2f B1 = bf[32];
        v2f B2 = bf[64];
        v2f B3 = bf[96];

        acc0 = __builtin_amdgcn_wmma_f32_16x16x4_f32(false, A, false, B0, (short)0, acc0, false, false);
        acc1 = __builtin_amdgcn_wmma_f32_16x16x4_f32(false, A, false, B1, (short)0, acc1, false, false);
        acc2 = __builtin_amdgcn_wmma_f32_16x16x4_f32(false, A, false, B2, (short)0, acc2, false, false);
        acc3 = __builtin_amdgcn_wmma_f32_16x16x4_f32(false, A, false, B3, (short)0, acc3, false, false);
    }

    // ---- add lin_b, store per C layout (VGPR i -> rows i / i+8) ----
    const int ncol = lane & 15;
    const float lb0 = sLB[ncol +  0];
    const float lb1 = sLB[ncol + 16];
    const float lb2 = sLB[ncol + 32];
    const float lb3 = sLB[ncol + 48];
    const int mbase = (lane < 16) ? 0 : 8;
    float* outp = out + (long)m0 * PATCH + ncol;

#pragma unroll
    for (int i = 0; i < 8; ++i) {
        float* row = outp + (mbase + i) * PATCH;
        row[0]  = acc0[i] + lb0;
        row[16] = acc1[i] + lb1;
        row[32] = acc2[i] + lb2;
        row[48] = acc3[i] + lb3;
    }
}

extern "C" void kernel_launch(void* const* d_in, const int* in_sizes, int n_in,
                              void* d_out, int out_size, void* d_ws, size_t ws_size,
                              hipStream_t stream) {
    const float* x      = (const float*)d_in[0];
    const float* conv_w = (const float*)d_in[1];
    const float* conv_b = (const float*)d_in[2];
    const float* lin_w  = (const float*)d_in[3];
    const float* lin_b  = (const float*)d_in[4];
    float* out = (float*)d_out;

    // 65536 M-tiles of 16 rows, 8 waves (tiles) per 256-thread block
    dim3 grid(8192), block(256);
    hipLaunchKernelGGL(fused_conv_gemm, grid, block, 0, stream,
                       x, conv_w, conv_b, lin_w, lin_b, out);
}